// Q_NETWORK_60292750901677
// MI455X (gfx1250) — compile-verified
//
#include <hip/hip_runtime.h>

#define N_NODES 250000
#define N_EDGES 1000000
#define DEG 16
#define EDIM 10
#define DD 14
#define ALPHA 0.3f

typedef __attribute__((ext_vector_type(2))) float v2f;
typedef __attribute__((ext_vector_type(8))) float v8f;

__device__ __forceinline__ float fast_rcp(float x) { return __builtin_amdgcn_rcpf(x); }
__device__ __forceinline__ float sigf(float x) { return fast_rcp(1.f + __expf(-x)); }
__device__ __forceinline__ float tanh_fast(float x) {
  // tanh(x) = 1 - 2/(1+exp(2x));  exp overflow -> inf -> rcp -> 0 -> +1; underflow -> -1
  return 1.f - 2.f * fast_rcp(1.f + __expf(2.f * x));
}

__device__ __forceinline__ v8f wmma4(v2f a, v2f b, v8f c) {
  // D(16x16,f32) = A(16x4,f32) x B(4x16,f32) + C
  return __builtin_amdgcn_wmma_f32_16x16x4_f32(false, a, false, b, (short)0, c, false, false);
}

// C[16x16] += A[16xKPAD] (LDS, row-major, lda) x B[KPADx16] (LDS, row-major, ldb, col offset nb)
// A frag: lane m=lane&15 holds K = k0+kh, k0+kh+1 with kh = 2*(lane>>4)
// B frag: lane n=lane&15 holds rows k0+kh, k0+kh+1
template <int KPAD>
__device__ __forceinline__ v8f gemm16(const float* A, int lda, const float* B, int ldb,
                                      int nb, int lane, v8f acc) {
  const int m = lane & 15;
  const int kh = (lane >> 4) * 2;
  const int n = lane & 15;
#pragma unroll
  for (int k0 = 0; k0 < KPAD; k0 += 4) {
    v2f a, b;
    a.x = A[m * lda + k0 + kh];
    a.y = A[m * lda + k0 + kh + 1];
    b.x = B[(k0 + kh) * ldb + nb + n];
    b.y = B[(k0 + kh + 1) * ldb + nb + n];
    acc = wmma4(a, b, acc);
  }
  return acc;
}

__global__ void zero_kernel(float* p, int n) {
  int i = blockIdx.x * blockDim.x + threadIdx.x;
  if (i < n) p[i] = 0.f;
}

// ---------------- edge MLP + scatter-add ----------------
// e = relu(sigmoid(ef@W1+b1)@W2+b2); seg[eidx[e]] += e    (16 edges per wave)
__global__ void __launch_bounds__(256) edge_kernel(
    const float* __restrict__ ef, const float* __restrict__ W1, const float* __restrict__ b1,
    const float* __restrict__ W2, const float* __restrict__ b2, const int* __restrict__ eidx,
    float* __restrict__ seg) {
  __shared__ float sW1[12 * 16], sW2[12 * 16], sb1[16], sb2[16];
  __shared__ float sA[8][16 * 12];
  __shared__ float sH[8][16 * 12];
  const int tid = threadIdx.x, lane = tid & 31, w = tid >> 5;

  for (int i = tid; i < 12 * 16; i += 256) {
    int k = i >> 4, n = i & 15;
    bool in = (k < EDIM) && (n < EDIM);
    sW1[i] = in ? W1[k * EDIM + n] : 0.f;
    sW2[i] = in ? W2[k * EDIM + n] : 0.f;
  }
  if (tid < 16) {
    sb1[tid] = tid < EDIM ? b1[tid] : 0.f;
    sb2[tid] = tid < EDIM ? b2[tid] : 0.f;
  }
  const int base = (blockIdx.x * 8 + w) * 16;

  // prefetch the tile 16 blocks ahead of edge feats into cache (global_prefetch_b8)
  {
    int pe = base + 128 * 16;  // 16 grid-blocks (2048 edges) ahead for this wave slot
    if (pe + 15 < N_EDGES) __builtin_prefetch(&ef[(size_t)pe * EDIM + lane * 5], 0, 1);
  }

  for (int i = lane; i < 16 * 12; i += 32) { sA[w][i] = 0.f; sH[w][i] = 0.f; }
  for (int i = lane; i < 16 * EDIM; i += 32) {
    int r = i / EDIM, c = i % EDIM;
    int e = base + r;
    sA[w][r * 12 + c] = (e < N_EDGES) ? ef[(size_t)e * EDIM + c] : 0.f;
  }
  __syncthreads();

  const int col = lane & 15;
  const int rh = (lane >> 4) * 8;
  v8f acc = {};
  acc = gemm16<12>(sA[w], 12, sW1, 16, 0, lane, acc);
#pragma unroll
  for (int v = 0; v < 8; ++v) {
    float s = sigf(acc[v] + sb1[col]);
    if (col < EDIM) sH[w][(v + rh) * 12 + col] = s;
  }
  __syncthreads();

  v8f acc2 = {};
  acc2 = gemm16<12>(sH[w], 12, sW2, 16, 0, lane, acc2);
#pragma unroll
  for (int v = 0; v < 8; ++v) {
    int r = v + rh;
    int e = base + r;
    float val = fmaxf(acc2[v] + sb2[col], 0.f);
    if (col < EDIM && e < N_EDGES)
      atomicAdd(&seg[(size_t)eidx[e] * EDIM + col], val);
  }
}

// ---------------- node: x-assembly, Wn, LSTM, si/sj ----------------
__global__ void __launch_bounds__(256) node_kernel(
    const float* __restrict__ seg, const int* __restrict__ phase,
    const float* __restrict__ Wn, const float* __restrict__ bn,
    const float* __restrict__ lk, const float* __restrict__ lrk, const float* __restrict__ lb,
    const float* __restrict__ h0, const float* __restrict__ c0, const float* __restrict__ Wa,
    float* __restrict__ node_vec, float* __restrict__ h_out, float* __restrict__ c_out,
    float* __restrict__ si, float* __restrict__ sj) {
  __shared__ float sWn[16 * 16], sbn[16], sK[16 * 64], sRK[16 * 64], sb[64], sWa[28];
  __shared__ float sX[8][16 * 16];
  __shared__ float sH0[8][16 * 16];
  __shared__ float sV[8][16 * 16];
  __shared__ float sZ[8][16 * 56];
  __shared__ float sHh[8][16 * 16];
  const int tid = threadIdx.x, lane = tid & 31, w = tid >> 5;

  for (int i = tid; i < 16 * 16; i += 256) {
    int k = i >> 4, n = i & 15;
    sWn[i] = (k < DD && n < DD) ? Wn[k * DD + n] : 0.f;
  }
  for (int i = tid; i < 16 * 64; i += 256) {
    int k = i >> 6, n = i & 63;
    bool in = (k < DD) && (n < 4 * DD);
    sK[i]  = in ? lk[k * 4 * DD + n] : 0.f;
    sRK[i] = in ? lrk[k * 4 * DD + n] : 0.f;
  }
  if (tid < 64) sb[tid] = tid < 4 * DD ? lb[tid] : 0.f;
  if (tid < 16) sbn[tid] = tid < DD ? bn[tid] : 0.f;
  if (tid < 28) sWa[tid] = Wa[tid];

  const int base = (blockIdx.x * 8 + w) * 16;
  for (int i = lane; i < 256; i += 32) {
    int r = i >> 4, c = i & 15;
    int nd = base + r;
    float x = 0.f, hh0 = 0.f;
    if (nd < N_NODES) {
      if (c < EDIM) x = seg[(size_t)nd * EDIM + c];
      if (c == EDIM - 1 + phase[nd]) x += 1.f;   // one-hot at col 9+phase
      if (c < DD) hh0 = h0[(size_t)nd * DD + c];
    }
    sX[w][i] = x;
    sH0[w][i] = hh0;
  }
  __syncthreads();

  const int col = lane & 15;
  const int rh = (lane >> 4) * 8;
  {
    v8f acc = {};
    acc = gemm16<16>(sX[w], 16, sWn, 16, 0, lane, acc);
#pragma unroll
    for (int v = 0; v < 8; ++v) {
      int r = v + rh;
      int nd = base + r;
      float val = acc[v] + sbn[col];      // cols >= 14: 0 + 0
      sV[w][r * 16 + col] = val;
      if (nd < N_NODES && col < DD) node_vec[(size_t)nd * DD + col] = val;
    }
  }
  __syncthreads();

#pragma unroll
  for (int t = 0; t < 4; ++t) {
    v8f acc = {};
    acc = gemm16<16>(sV[w], 16, sK, 64, t * 16, lane, acc);
    acc = gemm16<16>(sH0[w], 16, sRK, 64, t * 16, lane, acc);
    int cg = t * 16 + col;
#pragma unroll
    for (int v = 0; v < 8; ++v) {
      int r = v + rh;
      if (cg < 4 * DD) sZ[w][r * 56 + cg] = acc[v] + sb[cg];
    }
  }
  __syncthreads();

  for (int i = lane; i < 16 * DD; i += 32) {
    int r = i / DD, d = i % DD;
    int nd = base + r;
    float zi = sZ[w][r * 56 + d];
    float zf = sZ[w][r * 56 + DD + d];
    float zg = sZ[w][r * 56 + 2 * DD + d];
    float zo = sZ[w][r * 56 + 3 * DD + d];
    float cp = (nd < N_NODES) ? c0[(size_t)nd * DD + d] : 0.f;
    float cc = sigf(zf) * cp + sigf(zi) * tanh_fast(zg);
    float hh = sigf(zo) * tanh_fast(cc);
    sHh[w][r * 16 + d] = hh;
    if (nd < N_NODES) {
      c_out[(size_t)nd * DD + d] = cc;
      h_out[(size_t)nd * DD + d] = hh;
    }
  }
  __syncthreads();

  if (lane < 16) {
    int r = lane, nd = base + r;
    float a = 0.f, b = 0.f;
    for (int d = 0; d < DD; ++d) {
      float hh = sHh[w][r * 16 + d];
      a += hh * sWa[d];
      b += hh * sWa[DD + d];
    }
    if (nd < N_NODES) { si[nd] = a; sj[nd] = b; }
  }
}

// ---------------- attention (16 contiguous neighbors) + q-head ----------------
__global__ void __launch_bounds__(256) attn_q_kernel(
    const float* __restrict__ node_vec, const float* __restrict__ h,
    const float* __restrict__ si, const float* __restrict__ sj,
    const int* __restrict__ nsrc,
    const float* __restrict__ Wq1, const float* __restrict__ bq1,
    const float* __restrict__ Wq2, const float* __restrict__ bq2,
    float* __restrict__ q) {
  __shared__ float sW[44 * 32], sb1[32], sW2[60], sb2[2];
  __shared__ float sP[8][2 * 16 * 15];   // [half][row][denom, agg0..13]
  __shared__ float sA[8][16 * 44];
  __shared__ float sT[8][16 * 32];
  const int tid = threadIdx.x, lane = tid & 31, w = tid >> 5;

  for (int i = tid; i < 44 * 32; i += 256) {
    int k = i >> 5, n = i & 31;
    sW[i] = (k < 42 && n < 30) ? Wq1[k * 30 + n] : 0.f;
  }
  if (tid < 32) sb1[tid] = tid < 30 ? bq1[tid] : 0.f;
  if (tid < 60) sW2[tid] = Wq2[tid];
  if (tid < 2) sb2[tid] = bq2[tid];

  const int base = (blockIdx.x * 8 + w) * 16;
  {
    int r = lane & 15, half = lane >> 4;
    int nd = base + r;
    float pd = 0.f;
    float pa[DD];
#pragma unroll
    for (int d = 0; d < DD; ++d) pa[d] = 0.f;
    if (nd < N_NODES) {
      float s_i = si[nd];
      for (int j = 0; j < 8; ++j) {
        int src = nsrc[(size_t)nd * DEG + half * 8 + j];
        float s = s_i + sj[src];
        s = s > 0.f ? s : ALPHA * s;
        float wt = __expf(s);
        pd += wt;
        const float* hp = h + (size_t)src * DD;
#pragma unroll
        for (int d = 0; d < DD; ++d) pa[d] += wt * hp[d];
      }
    }
    float* p = &sP[w][(half * 16 + r) * 15];
    p[0] = pd;
#pragma unroll
    for (int d = 0; d < DD; ++d) p[1 + d] = pa[d];
  }
  __syncthreads();

  for (int i = lane; i < 16 * 44; i += 32) {
    int r = i / 44, c = i % 44;
    int nd = base + r;
    float val = 0.f;
    if (nd < N_NODES) {
      if (c < DD) val = node_vec[(size_t)nd * DD + c];
      else if (c < 2 * DD) val = h[(size_t)nd * DD + (c - DD)];
      else if (c < 3 * DD) {
        int d = c - 2 * DD;
        const float* p0 = &sP[w][r * 15];
        const float* p1 = &sP[w][(16 + r) * 15];
        float rden = fast_rcp(p0[0] + p1[0]);
        val = (p0[1 + d] + p1[1 + d]) * rden;
      }
    }
    sA[w][i] = val;
  }
  __syncthreads();

  const int col = lane & 15;
  const int rh = (lane >> 4) * 8;
#pragma unroll
  for (int t = 0; t < 2; ++t) {
    v8f acc = {};
    acc = gemm16<44>(sA[w], 44, sW, 32, t * 16, lane, acc);
    int cg = t * 16 + col;
#pragma unroll
    for (int v = 0; v < 8; ++v) {
      int r = v + rh;
      sT[w][r * 32 + cg] = fmaxf(acc[v] + sb1[cg], 0.f);
    }
  }
  __syncthreads();

  if (lane < 16) {
    int r = lane, nd = base + r;
    float q0 = sb2[0], q1 = sb2[1];
    for (int j = 0; j < 30; ++j) {
      float tv = sT[w][r * 32 + j];
      q0 += tv * sW2[j * 2];
      q1 += tv * sW2[j * 2 + 1];
    }
    if (nd < N_NODES) {
      q[(size_t)nd * 2] = fmaxf(q0, 0.f);
      q[(size_t)nd * 2 + 1] = fmaxf(q1, 0.f);
    }
  }
}

extern "C" void kernel_launch(void* const* d_in, const int* in_sizes, int n_in,
                              void* d_out, int out_size, void* d_ws, size_t ws_size,
                              hipStream_t stream) {
  const float* ef  = (const float*)d_in[0];
  const float* h0  = (const float*)d_in[1];
  const float* c0  = (const float*)d_in[2];
  const float* W1  = (const float*)d_in[3];
  const float* b1  = (const float*)d_in[4];
  const float* W2  = (const float*)d_in[5];
  const float* b2  = (const float*)d_in[6];
  const float* Wn  = (const float*)d_in[7];
  const float* bn  = (const float*)d_in[8];
  const float* lk  = (const float*)d_in[9];
  const float* lrk = (const float*)d_in[10];
  const float* lb  = (const float*)d_in[11];
  const float* Wa  = (const float*)d_in[12];
  const float* Wq1 = (const float*)d_in[13];
  const float* bq1 = (const float*)d_in[14];
  const float* Wq2 = (const float*)d_in[15];
  const float* bq2 = (const float*)d_in[16];
  const int* eidx  = (const int*)d_in[17];
  const int* phase = (const int*)d_in[18];
  const int* nsrc  = (const int*)d_in[19];
  // d_in[20] (nbr_dst) unused: it is repeat(arange(N), DEG) -> contiguous segments

  float* out  = (float*)d_out;
  float* qout = out;                              // [N,2]
  float* hout = out + (size_t)N_NODES * 2;        // [N,14]
  float* cout = out + (size_t)N_NODES * 16;       // [N,14]

  float* ws       = (float*)d_ws;
  float* seg      = ws;                           // N*10
  float* node_vec = ws + (size_t)N_NODES * 10;    // N*14
  float* si       = ws + (size_t)N_NODES * 24;    // N
  float* sj       = ws + (size_t)N_NODES * 25;    // N

  const int nseg = N_NODES * EDIM;
  zero_kernel<<<(nseg + 255) / 256, 256, 0, stream>>>(seg, nseg);
  edge_kernel<<<(N_EDGES + 127) / 128, 256, 0, stream>>>(ef, W1, b1, W2, b2, eidx, seg);
  node_kernel<<<(N_NODES + 127) / 128, 256, 0, stream>>>(seg, phase, Wn, bn, lk, lrk, lb,
                                                         h0, c0, Wa, node_vec, hout, cout, si, sj);
  attn_q_kernel<<<(N_NODES + 127) / 128, 256, 0, stream>>>(node_vec, hout, si, sj, nsrc,
                                                           Wq1, bq1, Wq2, bq2, qout);
}